// EdgeAttentionBias_40785009443410
// MI455X (gfx1250) — compile-verified
//
#include <hip/hip_runtime.h>
#include <hip/hip_bf16.h>

typedef __attribute__((ext_vector_type(2))) float v2f;
typedef __attribute__((ext_vector_type(8))) float v8f;

// ---------------------------------------------------------------------------
// Kernel 1: zero-fill the (H, N, N) output. 128 MiB of b128 stores — this is
// the dominant cost of the whole problem (~5.7 us at 23.3 TB/s HBM).
// ---------------------------------------------------------------------------
__global__ void EdgeBias_zero_kernel(float4* __restrict__ out, long long n4) {
    long long i = (long long)blockIdx.x * blockDim.x + threadIdx.x;
    long long stride = (long long)gridDim.x * blockDim.x;
    float4 z;
    z.x = 0.f; z.y = 0.f; z.z = 0.f; z.w = 0.f;
    for (; i < n4; i += stride) {
        out[i] = z;
    }
}

// ---------------------------------------------------------------------------
// Kernel 2: per wave, 16 edges per iteration through one
// V_WMMA_F32_16X16X4_F32:
//   A (16x4, f32):  row, col, score, 1.0   (lanes 0-15: K=0,1; lanes 16-31: K=2,3)
//   B (4x16, f32):  cols 0..7 = [W^T ; b], cols 8..15 = 0
//   D (16x16, f32): D[m][n] = logit(edge base+m, head n), n<8 valid
// Fast sigmoid (v_exp_f32 + v_rcp_f32), 32-bit scatter addressing.
// ---------------------------------------------------------------------------
__global__ void EdgeBias_wmma_scatter_kernel(const int*   __restrict__ edge_index, // [2, E]
                                             const float* __restrict__ edge_score, // [E]
                                             const float* __restrict__ W,          // [H=8, 3]
                                             const float* __restrict__ bvec,       // [H=8]
                                             const int*   __restrict__ num_nodes_p,
                                             float*       __restrict__ out,        // [8, N, N]
                                             int E_total) {
    const unsigned N  = (unsigned)(*num_nodes_p);
    const unsigned NN = N * N;                       // 4M for N=2048, fits u32

    const int lane = (int)(threadIdx.x & 31u);
    const int n    = lane & 15;      // head column (valid if n < 8)
    const int half = lane >> 4;      // K-half selector for A/B layout

    const int waves_per_block = blockDim.x >> 5;
    const int wave_id     = blockIdx.x * waves_per_block + (threadIdx.x >> 5);
    const int total_waves = gridDim.x * waves_per_block;

    // ---- Build B matrix (loop-invariant). VGPR0 = B[2*half][n], VGPR1 = B[2*half+1][n]
    float b0 = 0.0f, b1 = 0.0f;
    if (n < 8) {
        const int k0 = half * 2;     // 0 or 2
        b0 = (k0 < 3) ? W[n * 3 + k0] : bvec[n];
        const int k1 = k0 + 1;       // 1 or 3
        b1 = (k1 < 3) ? W[n * 3 + k1] : bvec[n];
    }
    v2f Bm; Bm[0] = b0; Bm[1] = b1;

    const unsigned headOff = (unsigned)n * NN;       // n*N^2 (garbage for n>=8, unused)
    const bool head_ok = (n < 8);

    const int num_tiles = (E_total + 15) >> 4;
    for (int tile = wave_id; tile < num_tiles; tile += total_waves) {
        const int base = tile << 4;
        int e = base + n;
        if (e >= E_total) e = E_total - 1;           // clamp (E is a multiple of 16 anyway)

        const int   r = edge_index[e];               // row id of edge base+n
        const int   c = edge_index[E_total + e];     // col id of edge base+n
        const float s = edge_score[e];

        // Branchless A fragment: lanes 0-15 carry (row, col); lanes 16-31 carry (score, 1.0)
        v2f Am;
        Am[0] = (half == 0) ? (float)r : s;
        Am[1] = (half == 0) ? (float)c : 1.0f;

        v8f acc = {};
        acc = __builtin_amdgcn_wmma_f32_16x16x4_f32(
            /*neg_a=*/false, Am, /*neg_b=*/false, Bm,
            /*c_mod=*/(short)0, acc, /*reuse_a=*/false, /*reuse_b=*/false);

        const int valid = E_total - base;            // edges remaining in this tile

        // D VGPR j holds edges m = j + 8*half for this lane; head = n.
        #pragma unroll
        for (int j = 0; j < 8; ++j) {
            const int m = j + half * 8;
            // Full-EXEC shuffles (ds_bpermute reads disabled lanes as zero, so
            // these must happen before any store predication).
            const unsigned rm = (unsigned)__shfl(r, m, 32);
            const unsigned cm = (unsigned)__shfl(c, m, 32);
            if (head_ok && m < valid) {
                // sigmoid(x) = 1 / (1 + exp(-x)) via fast v_exp_f32 + v_rcp_f32
                const float ex = __expf(-acc[j]);
                const float v  = __builtin_amdgcn_rcpf(1.0f + ex);
                // 32-bit flat offset (< 2^25 elements) -> SADDR-form store
                const unsigned off = headOff + rm * N + cm;
                out[off] = v;
            }
        }
    }
}

extern "C" void kernel_launch(void* const* d_in, const int* in_sizes, int n_in,
                              void* d_out, int out_size, void* d_ws, size_t ws_size,
                              hipStream_t stream) {
    const int*   edge_index = (const int*)  d_in[0];   // [2, E] int32
    const float* edge_score = (const float*)d_in[1];   // [E]
    const float* W          = (const float*)d_in[2];   // [8, 3]
    const float* bvec       = (const float*)d_in[3];   // [8]
    const int*   num_nodes  = (const int*)  d_in[4];   // scalar

    float* out = (float*)d_out;
    const int E = in_sizes[1];                         // 262144

    // 1) Zero-fill output: out_size floats (divisible by 4 -> float4 stores).
    const long long n4 = (long long)out_size / 4;
    EdgeBias_zero_kernel<<<4096, 256, 0, stream>>>((float4*)out, n4);

    // 2) WMMA logits + sigmoid + scatter. 8 waves/block, 1024 blocks = 8192 waves;
    //    E/16 = 16384 tiles -> 2 tiles per wave.
    EdgeBias_wmma_scatter_kernel<<<1024, 256, 0, stream>>>(
        edge_index, edge_score, W, bvec, num_nodes, out, E);
}